// FiniteModel_33045478375828
// MI455X (gfx1250) — compile-verified
//
#include <hip/hip_runtime.h>

// FiniteModel forward (convex / soft selection), fused:
//   scores = X @ Y^T - b ; W = softmax(50*scores) ; f_x = rowsum(W*scores) ; choice = W @ Y
// FP32 WMMA (V_WMMA_F32_16X16X4_F32) for both GEMMs; softmax fused in registers.

typedef float v2f __attribute__((ext_vector_type(2)));
typedef float v8f __attribute__((ext_vector_type(8)));

#define N_SAMPLES 32768
#define NUM_C     2048
#define NUM_D     16
#define C_TILES   (NUM_C / 16)        // 128
#define WAVES_PER_BLOCK 4
#define LDS_STRIDE 18                 // padded row stride (floats): even (8B align) + conflict-light

__global__ __launch_bounds__(WAVES_PER_BLOCK * 32)
void finite_model_fused(const float* __restrict__ X,
                        const float* __restrict__ Y,
                        const float* __restrict__ B,
                        float* __restrict__ out_choice,
                        float* __restrict__ out_fx)
{
    __shared__ float wbuf[WAVES_PER_BLOCK][16 * LDS_STRIDE];

    const int tid  = threadIdx.x;
    const int wave = tid >> 5;
    const int lane = tid & 31;
    const int col  = lane & 15;   // N index inside a 16x16 tile
    const int half = lane >> 4;   // 0: lanes 0-15 (rows 0-7), 1: lanes 16-31 (rows 8-15)
    const int nb   = blockIdx.x * WAVES_PER_BLOCK + wave;  // 16-row block index
    const int nbase = nb * 16;

    float* wl = &wbuf[wave][0];

    const float kscale = 50.0f * 1.4426950408889634f;  // temp * log2(e)

    // ---- X tile in A-operand layout (loop invariant) ----
    // A 16x4 f32 layout: lane L%16 = row M; VGPR v, half h -> K = v + 2*h.
    v2f xA[4];
    #pragma unroll
    for (int k = 0; k < 4; ++k) {
        xA[k] = *(const v2f*)(X + (nbase + col) * NUM_D + 4 * k + 2 * half);
    }

    // =========== Pass 1: row maxima of scores ===========
    float m[8];
    #pragma unroll
    for (int j = 0; j < 8; ++j) m[j] = -3.4e38f;

    for (int ct = 0; ct < C_TILES; ++ct) {
        const int c0 = ct * 16;
        if (ct + 1 < C_TILES)
            __builtin_prefetch(Y + (c0 + 16 + col) * NUM_D, 0, 1);

        const float bl = B[c0 + col];
        v8f acc;
        #pragma unroll
        for (int j = 0; j < 8; ++j) acc[j] = -bl;   // bias folded into C operand

        #pragma unroll
        for (int k = 0; k < 4; ++k) {
            // B1[K=d][N=c] = Y[c0+col][d], d = 4k + v + 2*half (v2f -> consecutive d)
            v2f yB = *(const v2f*)(Y + (c0 + col) * NUM_D + 4 * k + 2 * half);
            acc = __builtin_amdgcn_wmma_f32_16x16x4_f32(
                false, xA[k], false, yB, (short)0, acc, false, false);
        }
        #pragma unroll
        for (int j = 0; j < 8; ++j) m[j] = fmaxf(m[j], acc[j]);
    }
    // Butterfly max within each 16-lane half: every lane ends with the
    // true maxima of the 8 rows its half owns.
    #pragma unroll
    for (int s = 1; s <= 8; s <<= 1) {
        #pragma unroll
        for (int j = 0; j < 8; ++j)
            m[j] = fmaxf(m[j], __shfl_xor(m[j], s, 32));
    }

    // =========== Pass 2: softmax + choice GEMM ===========
    float ssum[8], fsum[8];
    v8f ch;
    #pragma unroll
    for (int j = 0; j < 8; ++j) { ssum[j] = 0.0f; fsum[j] = 0.0f; ch[j] = 0.0f; }

    for (int ct = 0; ct < C_TILES; ++ct) {
        const int c0 = ct * 16;
        if (ct + 1 < C_TILES)
            __builtin_prefetch(Y + (c0 + 16 + col) * NUM_D, 0, 1);

        const float bl = B[c0 + col];
        v8f acc;
        #pragma unroll
        for (int j = 0; j < 8; ++j) acc[j] = -bl;
        #pragma unroll
        for (int k = 0; k < 4; ++k) {
            v2f yB = *(const v2f*)(Y + (c0 + col) * NUM_D + 4 * k + 2 * half);
            acc = __builtin_amdgcn_wmma_f32_16x16x4_f32(
                false, xA[k], false, yB, (short)0, acc, false, false);
        }

        // Unnormalized weights (consistent per-row max), accumulate Z and f partials.
        float w[8];
        #pragma unroll
        for (int j = 0; j < 8; ++j) {
            w[j] = __builtin_amdgcn_exp2f(kscale * (acc[j] - m[j]));
            ssum[j] += w[j];
            fsum[j] += w[j] * acc[j];
        }

        // Lane-transpose W tile through LDS: C/D layout -> A-operand layout.
        #pragma unroll
        for (int j = 0; j < 8; ++j)
            wl[(j + 8 * half) * LDS_STRIDE + col] = w[j];
        __syncthreads();

        v2f wA[4];
        #pragma unroll
        for (int k = 0; k < 4; ++k) {
            const int base = col * LDS_STRIDE + 4 * k + 2 * half;
            wA[k][0] = wl[base + 0];
            wA[k][1] = wl[base + 1];
        }
        __syncthreads();

        // choice += W_tile @ Y_tile : B2[K=c][N=d] = Y[c0+4k+v+2*half][col]
        #pragma unroll
        for (int k = 0; k < 4; ++k) {
            v2f yB2;
            yB2[0] = Y[(c0 + 4 * k + 0 + 2 * half) * NUM_D + col];
            yB2[1] = Y[(c0 + 4 * k + 1 + 2 * half) * NUM_D + col];
            ch = __builtin_amdgcn_wmma_f32_16x16x4_f32(
                false, wA[k], false, yB2, (short)0, ch, false, false);
        }
    }

    // ---- Final reductions: Z and f across the 16 lanes of each half ----
    #pragma unroll
    for (int s = 1; s <= 8; s <<= 1) {
        #pragma unroll
        for (int j = 0; j < 8; ++j) {
            ssum[j] += __shfl_xor(ssum[j], s, 32);
            fsum[j] += __shfl_xor(fsum[j], s, 32);
        }
    }

    // choice: VGPR j holds row (j + 8*half), column `col`; ssum[j] is exactly Z of that row.
    #pragma unroll
    for (int j = 0; j < 8; ++j) {
        const int row = j + 8 * half;
        out_choice[(nbase + row) * NUM_D + col] = ch[j] / ssum[j];
    }
    // f_x: one lane per half writes each row.
    #pragma unroll
    for (int j = 0; j < 8; ++j) {
        if (col == j)
            out_fx[nbase + j + 8 * half] = fsum[j] / ssum[j];
    }
}

extern "C" void kernel_launch(void* const* d_in, const int* in_sizes, int n_in,
                              void* d_out, int out_size, void* d_ws, size_t ws_size,
                              hipStream_t stream) {
    const float* X = (const float*)d_in[0];   // [32768, 16]
    const float* Y = (const float*)d_in[1];   // [1, 2048, 16]
    const float* B = (const float*)d_in[2];   // [1, 2048]

    float* out_choice = (float*)d_out;                       // [32768, 16]
    float* out_fx     = (float*)d_out + N_SAMPLES * NUM_D;   // [32768]

    const int n_blocks = (N_SAMPLES / 16) / WAVES_PER_BLOCK; // 512
    finite_model_fused<<<n_blocks, WAVES_PER_BLOCK * 32, 0, stream>>>(
        X, Y, B, out_choice, out_fx);
}